// RBFFDDivergence_91173565759602
// MI455X (gfx1250) — compile-verified
//
#include <hip/hip_runtime.h>
#include <stdint.h>

// Problem constants (match reference): B=8, N=100000, D=3, M=32
constexpr int kM = 32;
constexpr int kD = 3;
constexpr int kB = 8;

#define WPB    8                    // waves per 256-thread block (wave32)
#define REGION 512                  // per-wave staging: 128B idx + 384B weights
#define BUFSZ  (WPB * REGION)       // 4 KB per buffer, double buffered -> 8 KB LDS

__global__ __launch_bounds__(256)
void rbffd_divergence_kernel(const float* __restrict__ fs,
                             const float* __restrict__ weights,
                             const int*   __restrict__ sidx,
                             float*       __restrict__ out,
                             int N)
{
    __shared__ unsigned char smem[2 * BUFSZ];

    const int lane = threadIdx.x & 31;
    const int wave = threadIdx.x >> 5;
    const int wid  = blockIdx.x * WPB + wave;
    const int nwaves = gridDim.x * WPB;

    unsigned char* reg0 = smem + wave * REGION;
    unsigned char* reg1 = smem + BUFSZ + wave * REGION;
    // AS3 -> flat aperture keeps the LDS byte offset in the low 32 bits.
    const unsigned l0 = (unsigned)(uintptr_t)reg0;
    const unsigned l1 = (unsigned)(uintptr_t)reg1;

    const unsigned long long idx_base = (unsigned long long)(uintptr_t)sidx;
    const unsigned long long w_base   = (unsigned long long)(uintptr_t)weights;

    // Stage node n's 32 int32 indices (128B) + 96 f32 weights (384B) into LDS
    // with 3 fully-populated async-to-LDS instructions. Index copy FIRST:
    // async loads complete in order, so a partial wait releases indices early.
    auto issue = [&](int n, unsigned lbase) {
        unsigned long long gi  = idx_base + (unsigned long long)n * (kM * 4)      + (unsigned)(lane * 4);
        unsigned long long gw  = w_base   + (unsigned long long)n * (kD * kM * 4) + (unsigned)(lane * 8);
        unsigned long long gw2 = w_base   + (unsigned long long)n * (kD * kM * 4) + 256u + (unsigned)(lane * 4);
        asm volatile("global_load_async_to_lds_b32 %0, %1, off"
                     :: "v"(lbase + (unsigned)(lane * 4)), "v"(gi) : "memory");
        asm volatile("global_load_async_to_lds_b64 %0, %1, off"
                     :: "v"(lbase + 128u + (unsigned)(lane * 8)), "v"(gw) : "memory");
        asm volatile("global_load_async_to_lds_b32 %0, %1, off"
                     :: "v"(lbase + 384u + (unsigned)(lane * 4)), "v"(gw2) : "memory");
    };

    if (wid < N) issue(wid, l0);

    const size_t planeN3 = (size_t)3 * (unsigned)N;   // fs batch stride (floats)

    for (int n = wid, p = 0; n < N; n += nwaves, p ^= 1) {
        const int nn = n + nwaves;
        const bool havenext = (nn < N);
        if (havenext) {
            issue(nn, p ? l0 : l1);                        // prefetch next node
            asm volatile("s_wait_asynccnt 5" ::: "memory"); // current idx staged
        } else {
            asm volatile("s_wait_asynccnt 2" ::: "memory");
        }

        const unsigned char* reg = p ? reg1 : reg0;
        const int idx = *(const int*)(reg + lane * 4);     // idx[n, lane]
        const float* fp = fs + (size_t)3 * (unsigned)idx;

        // Kick off the latency-critical random gathers (L2) before waiting
        // on the streamed weights.
        float f[kB][3];
        #pragma unroll
        for (int b = 0; b < kB; ++b) {
            const float* fb = fp + (size_t)b * planeN3;    // -> global_load_b96
            f[b][0] = fb[0]; f[b][1] = fb[1]; f[b][2] = fb[2];
        }

        if (havenext) { asm volatile("s_wait_asynccnt 3" ::: "memory"); }
        else          { asm volatile("s_wait_asynccnt 0" ::: "memory"); }

        const float w0 = *(const float*)(reg + 128 + lane * 4);   // w[n,0,lane]
        const float w1 = *(const float*)(reg + 256 + lane * 4);   // w[n,1,lane]
        const float w2 = *(const float*)(reg + 384 + lane * 4);   // w[n,2,lane]

        float acc[kB];
        #pragma unroll
        for (int b = 0; b < kB; ++b)
            acc[b] = fmaf(w0, f[b][0], fmaf(w1, f[b][1], w2 * f[b][2]));

        // Reduce each batch accumulator across the 32 stencil lanes (wave32).
        #pragma unroll
        for (int s = 16; s >= 1; s >>= 1) {
            #pragma unroll
            for (int b = 0; b < kB; ++b)
                acc[b] += __shfl_xor(acc[b], s, 32);
        }

        if (lane < kB)
            out[(size_t)lane * (unsigned)N + (unsigned)n] = acc[lane];
    }
}

extern "C" void kernel_launch(void* const* d_in, const int* in_sizes, int n_in,
                              void* d_out, int out_size, void* d_ws, size_t ws_size,
                              hipStream_t stream) {
    const float* fs      = (const float*)d_in[0];   // (B, N, D) f32
    const float* weights = (const float*)d_in[1];   // (N, D, M) f32
    const int*   sidx    = (const int*)  d_in[2];   // (N, M) int32
    float*       out     = (float*)d_out;           // (B, N) f32

    const int N = in_sizes[1] / (kD * kM);          // 100000

    // 1250 blocks x 8 waves = 10000 waves -> exactly 10 pipelined nodes/wave.
    rbffd_divergence_kernel<<<1250, 256, 0, stream>>>(fs, weights, sidx, out, N);
}